// hgypergraph_constructor_88484916232594
// MI455X (gfx1250) — compile-verified
//
#include <hip/hip_runtime.h>
#include <hip/hip_bf16.h>

#define NNODES 10000
#define DIM 64
#define KNN 10
#define ALPHA_F 3.0f

typedef __attribute__((ext_vector_type(2))) float v2f;
typedef __attribute__((ext_vector_type(8))) float v8f;

// ---------------------------------------------------------------------------
// Kernel A: t = tanh(3*(emb[idx] @ W^T + b)), sq[i] = sum_d t[i][d]^2
// grid 2500 x 256 threads; 4 rows per block, 64 threads per row.
// ---------------------------------------------------------------------------
__global__ __launch_bounds__(256) void embed_kernel(
    const int* __restrict__ idx, const float* __restrict__ emb,
    const float* __restrict__ w, const float* __restrict__ b,
    float* __restrict__ t, float* __restrict__ sq) {
  __shared__ float sW[DIM * DIM];
  __shared__ float sE[4][DIM];
  __shared__ float sRed[4][DIM];
  const int tid = threadIdx.x;
  for (int i = tid; i < DIM * DIM; i += 256) sW[i] = w[i];
  const int rl = tid >> 6;          // 0..3 row-in-block
  const int d = tid & 63;           // 0..63 output dim
  const int row = blockIdx.x * 4 + rl;
  const int src = idx[row];
  sE[rl][d] = emb[src * DIM + d];
  __syncthreads();
  float acc = b[d];
  #pragma unroll
  for (int k = 0; k < DIM; k++) acc += sE[rl][k] * sW[d * DIM + k];
  const float val = tanhf(ALPHA_F * acc);
  t[row * DIM + d] = val;
  sRed[rl][d] = val * val;
  __syncthreads();
  for (int s = 32; s > 0; s >>= 1) {
    if (d < s) sRed[rl][d] += sRed[rl][d + s];
    __syncthreads();
  }
  if (d == 0) sq[row] = sRed[rl][0];
}

// ---------------------------------------------------------------------------
// Kernel B: WMMA distance tiles + per-row top-10 (smallest sq_j - 2*dot,
// ties -> smaller j, matching lax.top_k on -d2). Writes per-row neighbor
// key = 10 indices sorted ascending.
// grid 625 x 128 threads (4 waves). 16 rows per workgroup.
// ---------------------------------------------------------------------------
__global__ __launch_bounds__(128) void knn_kernel(
    const float* __restrict__ t, const float* __restrict__ sq,
    unsigned* __restrict__ keys) {
  __shared__ float slab[4][16 * 17];       // per-wave 16x16 C tile, padded
  __shared__ float candD[4][16][KNN];
  __shared__ int   candI[4][16][KNN];

  const int wave = threadIdx.x >> 5;
  const int lane = threadIdx.x & 31;
  const int lmod = lane & 15;
  const int lhi  = lane >> 4;
  const int i0 = blockIdx.x * 16;

  // Preload A fragments: a[s] holds t[i0+lmod][4s + 2*lhi .. +1]  (K=64)
  v2f a[16];
  const float* arow = t + (size_t)(i0 + lmod) * DIM + 2 * lhi;
  #pragma unroll
  for (int s = 0; s < 16; s++) a[s] = *(const v2f*)(arow + 4 * s);

  // private top-10 (value, index), tracked worst element
  float td[KNN]; int ti[KNN];
  #pragma unroll
  for (int q = 0; q < KNN; q++) { td[q] = __builtin_inff(); ti[q] = 0x7FFFFFFF; }
  float wd = __builtin_inff(); int wi = 0x7FFFFFFF; int wslot = 0;

  float* myslab = slab[wave];
  for (int jt = wave; jt < NNODES / 16; jt += 4) {
    const int j0 = jt * 16;
    const float* brow = t + (size_t)(j0 + lmod) * DIM + 2 * lhi;

    // Prefetch next j-tile for this wave (global_prefetch_b8; uniform branch)
    if (jt + 4 < NNODES / 16) {
      __builtin_prefetch(brow + 64 * DIM, 0, 3);
      __builtin_prefetch(brow + 64 * DIM + 32, 0, 3);
    }

    // Batch all 16 B-fragment loads (one clause, one wait), then 16 WMMAs
    v2f bf[16];
    #pragma unroll
    for (int s = 0; s < 16; s++) bf[s] = *(const v2f*)(brow + 4 * s);

    v8f c = {};
    #pragma unroll
    for (int s = 0; s < 16; s++) {
      // D = A(16x4 f32) * B(4x16 f32) + C   -> v_wmma_f32_16x16x4_f32
      c = __builtin_amdgcn_wmma_f32_16x16x4_f32(
          false, a[s], false, bf[s], (short)0, c, false, false);
    }

    // C layout: vgpr r, lane l -> M = r + 8*(l>>4), N = l&15
    #pragma unroll
    for (int r = 0; r < 8; r++) myslab[(r + 8 * lhi) * 17 + lmod] = c[r];
    asm volatile("s_wait_dscnt 0" ::: "memory");  // intra-wave LDS RAW

    if (lane < 16) {
      #pragma unroll 1
      for (int col = 0; col < 16; col++) {
        const int j = j0 + col;
        const float v = __builtin_fmaf(-2.0f, myslab[lane * 17 + col], sq[j]);
        if (v < wd || (v == wd && j < wi)) {
          td[wslot] = v; ti[wslot] = j;
          wd = td[0]; wi = ti[0]; wslot = 0;
          #pragma unroll
          for (int q = 1; q < KNN; q++) {
            if (td[q] > wd || (td[q] == wd && ti[q] > wi)) {
              wd = td[q]; wi = ti[q]; wslot = q;
            }
          }
        }
      }
    }
  }

  if (lane < 16) {
    #pragma unroll
    for (int q = 0; q < KNN; q++) {
      candD[wave][lane][q] = td[q];
      candI[wave][lane][q] = ti[q];
    }
  }
  __syncthreads();

  if (wave == 0 && lane < 16) {
    float md[KNN]; int mi[KNN];
    #pragma unroll
    for (int q = 0; q < KNN; q++) { md[q] = __builtin_inff(); mi[q] = 0x7FFFFFFF; }
    float mwd = __builtin_inff(); int mwi = 0x7FFFFFFF; int mws = 0;
    for (int w2 = 0; w2 < 4; w2++) {
      #pragma unroll 1
      for (int q = 0; q < KNN; q++) {
        const float v = candD[w2][lane][q];
        const int   j = candI[w2][lane][q];
        if (v < mwd || (v == mwd && j < mwi)) {
          md[mws] = v; mi[mws] = j;
          mwd = md[0]; mwi = mi[0]; mws = 0;
          #pragma unroll
          for (int q2 = 1; q2 < KNN; q2++) {
            if (md[q2] > mwd || (md[q2] == mwd && mi[q2] > mwi)) {
              mwd = md[q2]; mwi = mi[q2]; mws = q2;
            }
          }
        }
      }
    }
    // key = neighbor indices sorted ascending (insertion sort of 10 ints)
    #pragma unroll 1
    for (int x = 1; x < KNN; x++) {
      const int kv = mi[x];
      int y = x - 1;
      while (y >= 0 && mi[y] > kv) { mi[y + 1] = mi[y]; y--; }
      mi[y + 1] = kv;
    }
    #pragma unroll
    for (int q = 0; q < KNN; q++)
      keys[(size_t)(i0 + lane) * KNN + q] = (unsigned)mi[q];
  }
}

// ---------------------------------------------------------------------------
// Kernel C1: is_rep[i] = 1 iff no j < i with identical key
// ---------------------------------------------------------------------------
__global__ __launch_bounds__(256) void isrep_kernel(
    const unsigned* __restrict__ keys, int* __restrict__ isrep) {
  const int i = blockIdx.x * 256 + threadIdx.x;
  if (i >= NNODES) return;
  const unsigned* ki = keys + (size_t)i * KNN;
  const unsigned k0 = ki[0];
  int rep = 1;
  for (int j = 0; j < i; j++) {
    const unsigned* kj = keys + (size_t)j * KNN;
    if (kj[0] != k0) continue;
    bool eq = true;
    #pragma unroll
    for (int q = 1; q < KNN; q++)
      if (kj[q] != ki[q]) { eq = false; break; }
    if (eq) { rep = 0; break; }
  }
  isrep[i] = rep;
}

// ---------------------------------------------------------------------------
// Kernel C2: rank[i] = #{ reps j : key_j >lex key_i }  (binary-row lex
// ascending == index-tuple lex descending, so this is the output column)
// ---------------------------------------------------------------------------
__global__ __launch_bounds__(256) void rank_kernel(
    const unsigned* __restrict__ keys, const int* __restrict__ isrep,
    int* __restrict__ rank) {
  const int i = blockIdx.x * 256 + threadIdx.x;
  if (i >= NNODES) return;
  if (!isrep[i]) { rank[i] = -1; return; }
  const unsigned* ki = keys + (size_t)i * KNN;
  int r = 0;
  for (int j = 0; j < NNODES; j++) {
    if (!isrep[j]) continue;
    const unsigned* kj = keys + (size_t)j * KNN;
    int gt = 0;
    #pragma unroll
    for (int q = 0; q < KNN; q++) {
      if (kj[q] != ki[q]) { gt = (kj[q] > ki[q]); break; }
    }
    r += gt;
  }
  rank[i] = r;
}

// ---------------------------------------------------------------------------
// Kernel D0: zero the 200MB bf16 output with 16B vector stores
// ---------------------------------------------------------------------------
__global__ __launch_bounds__(256) void zero_kernel(uint4* __restrict__ out,
                                                   long n16) {
  const long i = (long)blockIdx.x * 256 + threadIdx.x;
  if (i < n16) out[i] = make_uint4(0u, 0u, 0u, 0u);
}

// ---------------------------------------------------------------------------
// Kernel D1: scatter H[node, hyperedge] = 1 for each representative row
// ---------------------------------------------------------------------------
__global__ __launch_bounds__(256) void scatter_kernel(
    const unsigned* __restrict__ keys, const int* __restrict__ isrep,
    const int* __restrict__ rank, __hip_bfloat16* __restrict__ out) {
  const int i = blockIdx.x * 256 + threadIdx.x;
  if (i >= NNODES) return;
  if (!isrep[i]) return;
  const int r = rank[i];
  const __hip_bfloat16 one = __float2bfloat16(1.0f);
  #pragma unroll
  for (int q = 0; q < KNN; q++) {
    const unsigned a = keys[(size_t)i * KNN + q];
    out[(size_t)a * NNODES + r] = one;
  }
}

// ---------------------------------------------------------------------------
extern "C" void kernel_launch(void* const* d_in, const int* in_sizes, int n_in,
                              void* d_out, int out_size, void* d_ws, size_t ws_size,
                              hipStream_t stream) {
  const int*   idx = (const int*)d_in[0];
  const float* emb = (const float*)d_in[1];
  const float* w   = (const float*)d_in[2];
  const float* b   = (const float*)d_in[3];
  __hip_bfloat16* out = (__hip_bfloat16*)d_out;

  // workspace carve-out (~3.1 MB)
  float*    t_buf = (float*)d_ws;                       // NNODES*DIM f32
  float*    sq    = t_buf + (size_t)NNODES * DIM;       // NNODES f32
  unsigned* keys  = (unsigned*)(sq + NNODES);           // NNODES*KNN u32
  int*      isrep = (int*)(keys + (size_t)NNODES * KNN);
  int*      rank  = isrep + NNODES;

  embed_kernel<<<NNODES / 4, 256, 0, stream>>>(idx, emb, w, b, t_buf, sq);
  knn_kernel<<<NNODES / 16, 128, 0, stream>>>(t_buf, sq, keys);
  isrep_kernel<<<(NNODES + 255) / 256, 256, 0, stream>>>(keys, isrep);
  rank_kernel<<<(NNODES + 255) / 256, 256, 0, stream>>>(keys, isrep, rank);

  const long n16 = ((long)NNODES * NNODES * 2) / 16;    // 12.5M uint4 stores
  zero_kernel<<<(int)((n16 + 255) / 256), 256, 0, stream>>>((uint4*)out, n16);
  scatter_kernel<<<(NNODES + 255) / 256, 256, 0, stream>>>(keys, isrep, rank, out);
}